// StateSpaceSimulator_120259084383
// MI455X (gfx1250) — compile-verified
//
#include <hip/hip_runtime.h>

typedef __attribute__((ext_vector_type(2))) float v2f;
typedef __attribute__((ext_vector_type(4))) float v4f;
typedef __attribute__((ext_vector_type(8))) float v8f;

#define H_DIM 64

// CDNA5 hardware tanh (V_TANH_F32), branch-free fallback.
static __device__ __forceinline__ float fast_tanh(float x) {
#if __has_builtin(__builtin_amdgcn_tanhf)
    return __builtin_amdgcn_tanhf(x);
#else
    const float e = __builtin_amdgcn_exp2f(2.88539008177793f * x); // 2*log2(e)
    return 1.0f - 2.0f * __builtin_amdgcn_rcpf(e + 1.0f);
#endif
}

// One wave32 simulates 16 batch rows for all T steps.
//
// GEMM1 computed TRANSPOSED: hidden^T = W1^T @ feat^T via
// V_WMMA_F32_16X16X4_F32.  A = W1^T tiles (constant registers), B = feat^T
// (state x / input u kept directly in B-operand layout: lane l holds feat
// dims {2*(l/16), 2*(l/16)+1} of batch l%16).
//
// Bias trick: the unused K=6 pad slot of the u-block carries the constant
// pre-activation (b1 + theta_f rows of W1) as an A-row, with feature value
// 1.0 on the B side, so the accumulator starts from inline zero.
//
// Result lands batch=lane, h=(VGPR,half): dx = hidden@W2*scale is a pure
// in-lane FMA reduction over register-resident W2 + 4 xor-16 shuffles.
__global__ __launch_bounds__(128, 1) void ssm_wmma_kernel(
    const float* __restrict__ x0,
    const float* __restrict__ u,
    const float* __restrict__ theta_f,
    const float* __restrict__ theta_h,
    const float* __restrict__ tau_u,
    const float* __restrict__ tau_y,
    const float* __restrict__ W1,
    const float* __restrict__ b1,
    const float* __restrict__ W2,
    const float* __restrict__ Wh,
    float* __restrict__ out_x,
    float* __restrict__ out_y,
    int B, int T)
{
    const int lane   = threadIdx.x & 31;
    const int wave   = threadIdx.x >> 5;
    const int tile   = blockIdx.x * 4 + wave;
    const int batch0 = tile * 16;
    if (batch0 >= B) return;                  // wave-uniform: EXEC stays all-ones

    const int  col   = lane & 15;             // batch owned by this lane
    const bool hi    = lane >= 16;
    const int  k0    = hi ? 2 : 0;            // state dims this lane carries
    const int  hbase = hi ? 8 : 0;            // h offset of this lane's C half
    const int  b     = batch0 + col;

    const float scale = tau_u[0] / tau_y[0];
    const float th    = theta_h[0];

    // ---- W1^T tiles in A-operand layout (M = h rows 16mt..16mt+15, K = feat) ----
    // A layout: lane l -> M=col; VGPR v -> K = v + 2*(l/16).
    // Au covers K=4..7: rows 4,5 = u weights; K=6 (hi-lane .x) = bias row;
    // K=7 = zero pad.
    v2f Awx[4], Awu[4];
#pragma unroll
    for (int mt = 0; mt < 4; ++mt) {
        const int h = mt * 16 + col;
        const int r = hi ? 2 : 0;                        // feat rows 0..3 = x
        Awx[mt].x = W1[(r + 0) * H_DIM + h];
        Awx[mt].y = W1[(r + 1) * H_DIM + h];
        float c = b1[h];                                 // bias + theta_f part
#pragma unroll
        for (int i = 0; i < 5; ++i) c += theta_f[i] * W1[(6 + i) * H_DIM + h];
        Awu[mt].x = hi ? c    : W1[4 * H_DIM + h];
        Awu[mt].y = hi ? 0.0f : W1[5 * H_DIM + h];
    }

    // ---- W2 rows for this half, register-resident, pre-scaled ----
    // h = 16*mt + v + hbase  (matches C-layout register (mt, v) of this lane)
    v4f w2v[4][8];
#pragma unroll
    for (int mt = 0; mt < 4; ++mt)
#pragma unroll
        for (int v = 0; v < 8; ++v) {
            v4f w = *reinterpret_cast<const v4f*>(W2 + (mt * 16 + v + hbase) * 4);
            w2v[mt][v].x = w.x * scale;
            w2v[mt][v].y = w.y * scale;
            w2v[mt][v].z = w.z * scale;
            w2v[mt][v].w = w.w * scale;
        }

    const float wh0 = Wh[k0];
    const float wh1 = Wh[k0 + 1];

    // ---- state x in B-operand layout: lane holds dims (k0, k0+1) of batch col ----
    float xa0 = x0[(size_t)b * 4 + k0];
    float xa1 = x0[(size_t)b * 4 + k0 + 1];
    float pl0 = xa0 * wh0 + xa1 * wh1;
    float y   = pl0 + __shfl_xor(pl0, 16, 32) + th;      // y0 = h(x0), UNclamped

    const size_t uBase = (size_t)b * T * 2;
    const size_t xBase = (size_t)b * T * 4;
    const size_t yBase = (size_t)b * T;

    for (int t = 0; t < T; ++t) {
        // ---- emit pre-step carry (xs[t], ys[t]) ----
        v2f xs; xs.x = xa0; xs.y = xa1;
        *reinterpret_cast<v2f*>(out_x + xBase + (size_t)t * 4 + k0) = xs;
        out_y[yBase + t] = y;                 // lane pair writes same value: benign

        // ---- u_t into B-operand K-block (K=4,5 = u; K=6 = bias 1.0; K=7 = 0) ----
        v2f uv = *reinterpret_cast<const v2f*>(u + uBase + (size_t)t * 2);
        const size_t tpf = (t + 64 < T) ? (size_t)(t + 64) : (size_t)t;
        __builtin_prefetch(u + uBase + tpf * 2, 0, 1);
        v2f Bfu; Bfu.x = hi ? 1.0f : uv.x;
        Bfu.y = hi ? 0.0f : uv.y;
        v2f Bfx; Bfx.x = xa0; Bfx.y = xa1;

        // ---- hidden^T = tanh(W1^T feat^T + bias): 2 WMMAs per h tile, C=0 ----
        float act[4][8];
#pragma unroll
        for (int mt = 0; mt < 4; ++mt) {
            v8f c = {0.0f, 0.0f, 0.0f, 0.0f, 0.0f, 0.0f, 0.0f, 0.0f};
            c = __builtin_amdgcn_wmma_f32_16x16x4_f32(
                    false, Awu[mt], false, Bfu, (short)0, c, false, false);
            c = __builtin_amdgcn_wmma_f32_16x16x4_f32(
                    false, Awx[mt], false, Bfx, (short)0, c, false, false);
#pragma unroll
            for (int k = 0; k < 8; ++k) act[mt][k] = fast_tanh(c[k]);
        }

        // ---- dx = act @ W2 (scaled): in-lane over this half's 32 h values ----
        float p0 = 0.0f, p1 = 0.0f, p2 = 0.0f, p3 = 0.0f;
#pragma unroll
        for (int mt = 0; mt < 4; ++mt)
#pragma unroll
            for (int v = 0; v < 8; ++v) {
                const v4f  w = w2v[mt][v];
                const float a = act[mt][v];
                p0 += a * w.x; p1 += a * w.y; p2 += a * w.z; p3 += a * w.w;
            }
        // combine the two h-halves (lane pair holds the same batch)
        const float d0 = p0 + __shfl_xor(p0, 16, 32);
        const float d1 = p1 + __shfl_xor(p1, 16, 32);
        const float d2 = p2 + __shfl_xor(p2, 16, 32);
        const float d3 = p3 + __shfl_xor(p3, 16, 32);

        const float xr0 = xa0 + (hi ? d2 : d0);          // x_raw (pre-clamp)
        const float xr1 = xa1 + (hi ? d3 : d1);

        // y_{t+1} = clip(h(x_raw), 0, 10)
        const float pr   = xr0 * wh0 + xr1 * wh1;
        const float ysum = pr + __shfl_xor(pr, 16, 32) + th;
        y = fminf(fmaxf(ysum, 0.0f), 10.0f);

        // x_{t+1} = clip(x_raw): dims 0..2 -> [0,10], dim 3 unbounded
        xa0 = fminf(fmaxf(xr0, 0.0f), 10.0f);
        xa1 = hi ? xr1 : fminf(fmaxf(xr1, 0.0f), 10.0f);
    }
}

extern "C" void kernel_launch(void* const* d_in, const int* in_sizes, int n_in,
                              void* d_out, int out_size, void* d_ws, size_t ws_size,
                              hipStream_t stream) {
    const float* x0      = (const float*)d_in[0];
    const float* u       = (const float*)d_in[1];
    const float* theta_f = (const float*)d_in[2];
    const float* theta_h = (const float*)d_in[3];
    const float* tau_u   = (const float*)d_in[4];
    const float* tau_y   = (const float*)d_in[5];
    const float* W1      = (const float*)d_in[6];
    const float* b1      = (const float*)d_in[7];
    const float* W2      = (const float*)d_in[8];
    const float* Wh      = (const float*)d_in[9];

    const int B = in_sizes[0] / 4;            // x_0 is (B, 4)
    const int T = in_sizes[1] / (B * 2);      // u   is (B, T, 2)

    float* out_x = (float*)d_out;             // (B, T, 4)
    float* out_y = out_x + (size_t)B * T * 4; // (B, T, 1)

    const int waves  = (B + 15) / 16;         // one wave per 16 batch rows
    const int blocks = (waves + 3) / 4;       // 4 waves (128 threads) per block
    ssm_wmma_kernel<<<blocks, 128, 0, stream>>>(
        x0, u, theta_f, theta_h, tau_u, tau_y, W1, b1, W2, Wh,
        out_x, out_y, B, T);
}